// _multiCodebookQuantization_48421461295639
// MI455X (gfx1250) — compile-verified
//
#include <hip/hip_runtime.h>
#include <stdint.h>

// Shapes from the reference
#define NN 16
#define MM 4
#define DD 32
#define KK 256
#define HH 64
#define WW 64
#define HWSZ (HH * WW)        // 4096 pixels per (n,m)
#define CBPAD 34              // padded LDS row stride (floats): conflict-free, b64-aligned

typedef __attribute__((ext_vector_type(2))) float v2f;
typedef __attribute__((ext_vector_type(4))) float v4f;
typedef __attribute__((ext_vector_type(8))) float v8f;
typedef __attribute__((ext_vector_type(4))) unsigned int u32x4;
typedef __attribute__((ext_vector_type(4))) int i32x4;
typedef __attribute__((ext_vector_type(8))) int i32x8;

#define LN2F 0.69314718055994530942f

__global__ __launch_bounds__(256) void mcq_fused_kernel(
    const float* __restrict__ x,         // (N, M*D, H, W)
    const float* __restrict__ codebook,  // (M, K, D)
    float* __restrict__ sample_out,      // (N, M, H, W, K)
    float* __restrict__ code_out,        // (N, M, H, W) as float
    float* __restrict__ logit_out)       // (N, M, H, W, K)
{
    __shared__ float lds_cb[KK * CBPAD]; // codebook[m], TDM-padded rows (32 data + 2 pad dwords)
    __shared__ float lds_c2[KK];         // per-code squared norm

    const int nm  = blockIdx.x >> 5;     // n*M + m  (64 total)
    const int tb  = blockIdx.x & 31;     // pixel-tile block within (n,m)
    const int m   = nm & (MM - 1);
    const int tid = threadIdx.x;
    const int wid = tid >> 5;

    const float* cbm = codebook + (size_t)m * KK * DD;

    // ---- Stage codebook[m] (256x32 fp32) into padded LDS via the Tensor Data Mover ----
#if __has_builtin(__builtin_amdgcn_tensor_load_to_lds)
    if (wid == 0) {
        const uint64_t ga = (uint64_t)(uintptr_t)cbm;
        const uint32_t la = (uint32_t)(uintptr_t)lds_cb;   // flat shared addr: low 32 bits = LDS offset
        u32x4 g0;
        g0[0] = 1u;                                        // count=1, user descriptor
        g0[1] = la;                                        // lds_addr (bytes)
        g0[2] = (uint32_t)ga;                              // global_addr[31:0]
        g0[3] = (uint32_t)((ga >> 32) & 0x01FFFFFFu)       // global_addr[56:32]
              | 0x80000000u;                               // type=2 ("image") in bits[127:126]
        i32x8 g1;
        g1[0] = (int)((2u << 16)                           // data_size = 4B
              |       (1u << 20)                           // pad_enable
              |       (4u << 22)                           // pad_interval: every 32 dwords
              |       (1u << 25));                         // pad_amount: 2 dwords -> row stride 34
        g1[1] = (int)(32u << 16);                          // tensor_dim0 = 32 (D), low 16 bits
        g1[2] = (int)(256u << 16);                         // tensor_dim0 hi=0 | tensor_dim1 = 256 (K) lo16
        g1[3] = (int)(32u << 16);                          // tensor_dim1 hi=0 | tile_dim0 = 32
        g1[4] = (int)256u;                                 // tile_dim1 = 256, tile_dim2 = 0
        g1[5] = (int)32u;                                  // tensor_dim0_stride = 32 elements
        g1[6] = 0;                                         // stride0 hi | tensor_dim1_stride lo (unused)
        g1[7] = 0;
        const i32x4 z4 = {0, 0, 0, 0};                     // 2D tile: groups 2/3 unused
        const i32x8 z8 = {0, 0, 0, 0, 0, 0, 0, 0};         // 6-arg form: extra group, unused here
        __builtin_amdgcn_tensor_load_to_lds(g0, g1, z4, z4, z8, 0);
        __builtin_amdgcn_s_wait_tensorcnt(0);
    }
#else
    for (int i = tid; i < KK * DD / 4; i += 256) {
        float4 v = ((const float4*)cbm)[i];
        int e = i * 4, row = e >> 5, col = e & 31;
        float* p = &lds_cb[row * CBPAD + col];
        p[0] = v.x; p[1] = v.y; p[2] = v.z; p[3] = v.w;
    }
#endif
    __syncthreads();
    { // c2[k] = sum_d cb[k,d]^2   (one k per thread, K == blockDim)
        float s = 0.f;
        #pragma unroll
        for (int d = 0; d < DD; ++d) { float c = lds_cb[tid * CBPAD + d]; s += c * c; }
        lds_c2[tid] = s;
    }
    __syncthreads();

    const int lane   = tid & 31;
    const int sub    = lane & 15;        // pixel column within tile / A-row within tile
    const bool hi    = (lane >= 16);
    const int d0base = hi ? 2 : 0;       // WMMA f32 16x16x4: lanes 0-15 hold K=0,1; 16-31 hold K=2,3
    const int rbase  = hi ? 8 : 0;       // C layout: lanes 16-31 hold rows M=8..15
    const int p0     = tb * 128 + wid * 16;
    const int pixg   = nm * HWSZ + p0 + sub;

    // ---- Load B fragments (x slab for these 16 pixels, read-once -> NT) + partial |x|^2 ----
    const float* xnm = x + (size_t)nm * DD * HWSZ;
    v2f b[8];
    float x2p = 0.f;
    #pragma unroll
    for (int dd = 0; dd < 8; ++dd) {
        int d0 = dd * 4 + d0base;
        float bx = __builtin_nontemporal_load(&xnm[(size_t)d0 * HWSZ + p0 + sub]);
        float by = __builtin_nontemporal_load(&xnm[(size_t)(d0 + 1) * HWSZ + p0 + sub]);
        b[dd][0] = bx; b[dd][1] = by;
        x2p += bx * bx + by * by;
    }
    float x2 = x2p + __shfl_xor(x2p, 16, 32);   // lane pair covers all 32 d's of this pixel

    // ---- 16 k-tiles: 8 chained fp32 WMMAs each; log + Gumbel-argmax epilogue ----
    // Gumbel-max over logits == argmax(dist / Exp(1)); scale-invariant, so use -log2(u)
    // directly and a raw v_rcp_f32 instead of libm log/divide sequences.
    float best = -3.0e38f;
    int bestk  = 0;
    float* logit_base = logit_out + (size_t)pixg * KK + rbase;
    const uint32_t pixmix = (uint32_t)pixg * 0x9E3779B1u + 0xC2B2AE3Du;

    #pragma unroll
    for (int kt = 0; kt < 16; ++kt) {
        v8f acc = {0.f, 0.f, 0.f, 0.f, 0.f, 0.f, 0.f, 0.f};
        const int arow = (kt * 16 + sub) * CBPAD + d0base;
        #pragma unroll
        for (int dd = 0; dd < 8; ++dd) {
            v2f a;
            a[0] = lds_cb[arow + dd * 4];
            a[1] = lds_cb[arow + dd * 4 + 1];
            // D = A(16 codes x 4 d) * B(4 d x 16 pixels) + C, full fp32
            acc = __builtin_amdgcn_wmma_f32_16x16x4_f32(
                false, a, false, b[dd], (short)0, acc, false, false);
        }
        v8f lg;
        #pragma unroll
        for (int r = 0; r < 8; ++r) {
            int kk = kt * 16 + rbase + r;
            float dist = x2 + lds_c2[kk] - 2.0f * acc[r];
            // natural log via raw v_log_f32 (log2) * ln2
            lg[r] = __builtin_amdgcn_logf(dist) * LN2F;
            // 2-round multiply-xorshift hash -> u in [0,1) as 23-bit mantissa splice
            uint32_t h = pixmix ^ ((uint32_t)kk * 0x85EBCA77u);
            h *= 0x7FEB352Du; h ^= h >> 15;
            h *= 0x846CA68Bu;
            float u = __uint_as_float(0x3F800000u | (h >> 9)) - 1.0f;  // [0,1)
            float w = -__builtin_amdgcn_logf(u);          // -log2(u) ~ scaled Exp(1); u==0 -> +inf -> z=0
            float z = dist * __builtin_amdgcn_rcpf(w);
            if (z > best) { best = z; bestk = kk; }
        }
        // coalesced logit store: lane pair writes one full 64B line per tile
        v4f s0 = {lg[0], lg[1], lg[2], lg[3]};
        v4f s1 = {lg[4], lg[5], lg[6], lg[7]};
        __builtin_nontemporal_store(s0, (v4f*)(logit_base + kt * 16));
        __builtin_nontemporal_store(s1, (v4f*)(logit_base + kt * 16 + 4));
    }

    // ---- combine argmax across the lane pair ----
    {
        float ob = __shfl_xor(best, 16, 32);
        int obk  = __shfl_xor(bestk, 16, 32);
        if (ob > best || (ob == best && obk < bestk)) { best = ob; bestk = obk; }
    }

    if (!hi) code_out[pixg] = (float)bestk;

    // ---- sample: straight-through forward value == one-hot(bestk) (exact for 0s, 1 ulp for the 1) ----
    float* samp_pix  = sample_out + (size_t)pixg * KK;
    float* samp_base = samp_pix + rbase;
    const v4f zero4 = {0.f, 0.f, 0.f, 0.f};
    #pragma unroll
    for (int kt = 0; kt < 16; ++kt) {
        __builtin_nontemporal_store(zero4, (v4f*)(samp_base + kt * 16));
        __builtin_nontemporal_store(zero4, (v4f*)(samp_base + kt * 16 + 4));
    }
    // fix-up: the lane that wrote bestk's 32B chunk overwrites one float with 1.0
    // (same-lane store->store ordering is guaranteed by the memory model)
    if (((bestk >> 3) & 1) == (hi ? 1 : 0)) {
        __builtin_nontemporal_store(1.0f, samp_pix + bestk);
    }
}

extern "C" void kernel_launch(void* const* d_in, const int* in_sizes, int n_in,
                              void* d_out, int out_size, void* d_ws, size_t ws_size,
                              hipStream_t stream) {
    const float* x  = (const float*)d_in[0];        // 16*128*64*64
    const float* cb = (const float*)d_in[1];        // 4*256*32
    float* out = (float*)d_out;
    float* sample_out = out;                                  // 67,108,864 elems
    float* code_out   = out + (size_t)NN * MM * HWSZ * KK;    // 262,144 elems
    float* logit_out  = code_out + (size_t)NN * MM * HWSZ;    // 67,108,864 elems

    dim3 grid(64 * 32);   // (n,m) pairs * 32 pixel-tile blocks
    dim3 block(256);      // 8 waves, one 16-pixel tile each
    hipLaunchKernelGGL(mcq_fused_kernel, grid, block, 0, stream,
                       x, cb, sample_out, code_out, logit_out);
}